// Hand2GripperModel_60825326846773
// MI455X (gfx1250) — compile-verified
//
#include <hip/hip_runtime.h>

// ---------------------------------------------------------------------------
// Fused GAT layer for 21-joint hand graphs on gfx1250 (MI455X).
//   Wh = h @ W             -> V_WMMA_F32_16X16X4_F32, Wh kept in LDS
//   attention + softmax    -> VALU per wave (tiny)
//   out = alpha @ Wh       -> V_WMMA_F32_16X16X4_F32 with zero-padded alpha
// Memory-bound kernel: only h (352MB) read + out (88MB) written; fusion avoids
// the 176MB Wh round-trip.
// ---------------------------------------------------------------------------

typedef __attribute__((ext_vector_type(2))) float v2f;
typedef __attribute__((ext_vector_type(8))) float v8f;

#define NJ      21
#define IN_DIM  256
#define OUT_DIM 64
#define BATCH   16384
#define BPB     16                 // batches per block
#define M_BLOCK (NJ * BPB)         // 336 rows = 21 tiles of 16 exactly
#define WT_S    260                // W^T row stride: 260%64==4 -> b64 loads hit 64 distinct banks
#define WH_S    68                 // Wh row stride: 8*68%64==32 -> conflict-free D stores
#define WH_ROWS 340                // 336 + 4 zero pad rows (read by padded alpha K=24)
#define AL_S    24                 // alpha row stride (K padded 21 -> 24)

// LDS layout (in floats)
#define OFF_WT  0
#define SZ_WT   (OUT_DIM * WT_S)        // 16640
#define OFF_WH  (OFF_WT + SZ_WT)
#define SZ_WH   (WH_ROWS * WH_S)        // 23120
#define OFF_AL  (OFF_WH + SZ_WH)
#define SZ_AL   (8 * 32 * AL_S)         // 6144 (per-wave 32x24 alpha)
#define OFF_SD  (OFF_AL + SZ_AL)
#define SZ_SD   (8 * 32)                // per-wave s_dst scratch
#define OFF_ADJ (OFF_SD + SZ_SD)
#define SZ_ADJ  (NJ * 22)               // 462
#define OFF_AV  (OFF_ADJ + SZ_ADJ)
#define SZ_AV   (2 * OUT_DIM)           // 128
#define SMEM_FLOATS (OFF_AV + SZ_AV)    // 46750 floats = 187000 bytes

__global__ __launch_bounds__(256) void gat_fused_kernel(
    const float* __restrict__ h,    // (B, 21, 256)
    const float* __restrict__ adj,  // (21, 21)
    const float* __restrict__ W,    // (256, 64)
    const float* __restrict__ a,    // (128,)
    float* __restrict__ out)        // (B, 21, 64)
{
    extern __shared__ float smem[];
    float* Wt  = smem + OFF_WT;     // W transposed: Wt[o*WT_S + k]
    float* Wh  = smem + OFF_WH;     // Wh[row*WH_S + o], row = 21*bl + n
    float* AL  = smem + OFF_AL;
    float* SD  = smem + OFF_SD;
    float* ADJ = smem + OFF_ADJ;
    float* AV  = smem + OFF_AV;

    const int tid  = threadIdx.x;
    const int wave = tid >> 5;
    const int lane = tid & 31;
    const int lo   = lane & 15;     // A/B/C lane column within half-wave
    const int hi   = lane >> 4;     // half-wave select (K pair / row+8)

    // ---------------- init: stage W^T, adj, a; zero Wh pad rows --------------
    for (int i = tid; i < IN_DIM * OUT_DIM; i += 256) {
        int k = i >> 6, o = i & 63;          // coalesced read of row-major W
        Wt[o * WT_S + k] = W[i];
    }
    for (int i = tid; i < NJ * NJ; i += 256) {
        int r = i / NJ, c = i - r * NJ;
        ADJ[r * 22 + c] = adj[i];
    }
    if (tid < 2 * OUT_DIM) AV[tid] = a[tid];
    for (int i = tid; i < (WH_ROWS - M_BLOCK) * WH_S; i += 256)
        Wh[M_BLOCK * WH_S + i] = 0.0f;
    __syncthreads();

    // ---------------- phase 1: Wh = h @ W via v_wmma_f32_16x16x4_f32 ---------
    const long row0 = (long)blockIdx.x * M_BLOCK;
    for (int mt = wave; mt < M_BLOCK / 16; mt += 8) {
        // A fragment source: lane (lo) owns row r0+lo, hi selects K pair {0,1}/{2,3}
        const float* hp = h + (row0 + mt * 16 + lo) * (long)IN_DIM + 2 * hi;
        // speculative prefetch of this wave's next M-tile (dropped if OOB)
        __builtin_prefetch(hp + 128 * IN_DIM, 0, 0);

        v8f acc0 = {}, acc1 = {}, acc2 = {}, acc3 = {};
        for (int k = 0; k < IN_DIM; k += 4) {
            v2f af = *(const v2f*)(hp + k);   // A: {h[r][k+2hi], h[r][k+1+2hi]}
            // B: {W[k+2hi][n], W[k+1+2hi][n]} == contiguous in W^T -> one b64
            v2f b0 = *(const v2f*)(Wt + (0  + lo) * WT_S + k + 2 * hi);
            v2f b1 = *(const v2f*)(Wt + (16 + lo) * WT_S + k + 2 * hi);
            v2f b2 = *(const v2f*)(Wt + (32 + lo) * WT_S + k + 2 * hi);
            v2f b3 = *(const v2f*)(Wt + (48 + lo) * WT_S + k + 2 * hi);
            acc0 = __builtin_amdgcn_wmma_f32_16x16x4_f32(false, af, false, b0, (short)0, acc0, false, false);
            acc1 = __builtin_amdgcn_wmma_f32_16x16x4_f32(false, af, false, b1, (short)0, acc1, false, false);
            acc2 = __builtin_amdgcn_wmma_f32_16x16x4_f32(false, af, false, b2, (short)0, acc2, false, false);
            acc3 = __builtin_amdgcn_wmma_f32_16x16x4_f32(false, af, false, b3, (short)0, acc3, false, false);
        }
        // D layout: VGPR j, lane -> row (mt*16 + j + 8*hi), col (nt*16 + lo)
        const int r0l = mt * 16;
        #pragma unroll
        for (int j = 0; j < 8; ++j) {
            const int r = (r0l + j + 8 * hi) * WH_S + lo;
            Wh[r +  0] = acc0[j];
            Wh[r + 16] = acc1[j];
            Wh[r + 32] = acc2[j];
            Wh[r + 48] = acc3[j];
        }
    }
    __syncthreads();

    // ---------------- phase 2: attention per wave (2 batches each) -----------
    float* alw = AL + wave * 32 * AL_S;
    float* sdw = SD + wave * 32;
    for (int ib = 0; ib < 2; ++ib) {
        const int   bl  = wave * 2 + ib;
        const float* whb = Wh + bl * NJ * WH_S;

        // s_src / s_dst: lane n (<21) owns joint n
        float ssrc = 0.0f, sdst = 0.0f;
        if (lane < NJ) {
            const float* r = whb + lane * WH_S;
            #pragma unroll 16
            for (int c = 0; c < OUT_DIM; ++c) {
                float v = r[c];
                ssrc = __builtin_fmaf(v, AV[c], ssrc);
                sdst = __builtin_fmaf(v, AV[OUT_DIM + c], sdst);
            }
        }
        sdw[lane] = sdst;                      // cross-lane exchange via LDS
        __builtin_amdgcn_wave_barrier();       // keep store before loads

        // leaky-relu logits, adjacency mask, softmax over m; write padded alpha
        if (lane < NJ) {
            float ev[NJ];
            float emax = -__builtin_inff();
            #pragma unroll
            for (int m = 0; m < NJ; ++m) {
                float e;
                if (ADJ[lane * 22 + m] != 0.0f) {
                    float t = ssrc + sdw[m];
                    e = (t >= 0.0f) ? t : 0.2f * t;
                } else {
                    e = -__builtin_inff();
                }
                ev[m] = e;
                emax  = fmaxf(emax, e);
            }
            float s = 0.0f;
            #pragma unroll
            for (int m = 0; m < NJ; ++m) { float x = __expf(ev[m] - emax); ev[m] = x; s += x; }
            const float inv = 1.0f / s;
            #pragma unroll
            for (int m = 0; m < NJ; ++m) alw[lane * AL_S + m] = ev[m] * inv;
            alw[lane * AL_S + 21] = 0.0f;      // K padding
            alw[lane * AL_S + 22] = 0.0f;
            alw[lane * AL_S + 23] = 0.0f;
        } else {
            #pragma unroll
            for (int m = 0; m < AL_S; ++m) alw[lane * AL_S + m] = 0.0f;  // M padding rows
        }
        __builtin_amdgcn_wave_barrier();

        // out = alpha(32x24, zero padded) @ Wh(24x64) via WMMA, masked store
        float* outb = out + ((long)(blockIdx.x * BPB + bl)) * NJ * OUT_DIM;
        for (int mt = 0; mt < 2; ++mt) {
            v8f c0 = {}, c1 = {}, c2 = {}, c3 = {};
            #pragma unroll
            for (int k = 0; k < AL_S; k += 4) {
                v2f af = *(const v2f*)(alw + (mt * 16 + lo) * AL_S + k + 2 * hi);
                const float* br0 = whb + (k + 2 * hi) * WH_S;
                const float* br1 = br0 + WH_S;
                v2f b0, b1, b2, b3;
                b0.x = br0[lo +  0]; b0.y = br1[lo +  0];
                b1.x = br0[lo + 16]; b1.y = br1[lo + 16];
                b2.x = br0[lo + 32]; b2.y = br1[lo + 32];
                b3.x = br0[lo + 48]; b3.y = br1[lo + 48];
                c0 = __builtin_amdgcn_wmma_f32_16x16x4_f32(false, af, false, b0, (short)0, c0, false, false);
                c1 = __builtin_amdgcn_wmma_f32_16x16x4_f32(false, af, false, b1, (short)0, c1, false, false);
                c2 = __builtin_amdgcn_wmma_f32_16x16x4_f32(false, af, false, b2, (short)0, c2, false, false);
                c3 = __builtin_amdgcn_wmma_f32_16x16x4_f32(false, af, false, b3, (short)0, c3, false, false);
            }
            #pragma unroll
            for (int j = 0; j < 8; ++j) {
                const int row = mt * 16 + j + 8 * hi;
                if (row < NJ) {
                    float* o = outb + row * OUT_DIM + lo;
                    o[ 0] = c0[j];
                    o[16] = c1[j];
                    o[32] = c2[j];
                    o[48] = c3[j];
                }
            }
        }
    }
}

extern "C" void kernel_launch(void* const* d_in, const int* in_sizes, int n_in,
                              void* d_out, int out_size, void* d_ws, size_t ws_size,
                              hipStream_t stream) {
    const float* h   = (const float*)d_in[0];
    const float* adj = (const float*)d_in[1];
    const float* W   = (const float*)d_in[2];
    const float* a   = (const float*)d_in[3];
    float* out = (float*)d_out;

    dim3 grid(BATCH / BPB);   // 1024 blocks of 16 batches
    dim3 block(256);          // 8 wave32
    size_t smem = SMEM_FLOATS * sizeof(float);
    hipLaunchKernelGGL(gat_fused_kernel, grid, block, smem, stream, h, adj, W, a, out);
}